// QAttention_64115271795092
// MI455X (gfx1250) — compile-verified
//
#include <hip/hip_runtime.h>
#include <math.h>

// QAttention for MI455X (gfx1250), fp32 exact-path.
// Algebraic reduction: tr(Uq Xq Uq~ Uk Xk Uk~) = tr(Xq * (W Xk W~)), W = Uq~ Uk.
// Only k in {0,1} of the trace and V at t in {0,1} are consumed by the output,
// so total work collapses to: 64 complex 128x128x128 conjugations (WMMA f32),
// 1024 complex traces (streaming 64MB of inputs), and a 64MB scaled write.
// Bandwidth-bound: ~128MB of mandatory HBM traffic.

#define DDIM 128
#define DD   (DDIM * DDIM)   // 16384
#define TT   32
#define BBATCH 16
#define NM   32              // batch of matrices in phase 1 (t in {0,1} x 16 b)

typedef float v2f __attribute__((ext_vector_type(2)));
typedef float v8f __attribute__((ext_vector_type(8)));

__device__ __forceinline__ v8f wmma_f32(v2f a, v2f b, v8f c) {
    // V_WMMA_F32_16X16X4_F32 : D = A(16x4) * B(4x16) + C, exact fp32.
    // 8-arg form: (neg_a, A, neg_b, B, c_mod, C, reuse_a, reuse_b).
    // f32 WMMA only supports C-negate, so neg_a/neg_b must be false.
    return __builtin_amdgcn_wmma_f32_16x16x4_f32(false, a, false, b, (short)0, c,
                                                 false, false);
}

// ---------------------------------------------------------------------------
// W = Uq^H * Uk  (Wr, Wi), and de-interleave Uv -> (Uvr, Uvi).
// U layout is (D, D, 2) with channel-last real/imag.
__global__ void k_prep(const float* __restrict__ Uq, const float* __restrict__ Uk,
                       const float* __restrict__ Uv,
                       float* __restrict__ Wr, float* __restrict__ Wi,
                       float* __restrict__ Uvr, float* __restrict__ Uvi) {
    int idx = blockIdx.x * blockDim.x + threadIdx.x;   // 0..16383
    int i = idx >> 7, j = idx & 127;
    float wr = 0.f, wi = 0.f;
    for (int m = 0; m < DDIM; ++m) {
        float qr = Uq[(m * DDIM + i) * 2 + 0];
        float qi = Uq[(m * DDIM + i) * 2 + 1];
        float kr = Uk[(m * DDIM + j) * 2 + 0];
        float ki = Uk[(m * DDIM + j) * 2 + 1];
        wr += qr * kr + qi * ki;      // Re( conj(Uq[m,i]) * Uk[m,j] )
        wi += qr * ki - qi * kr;      // Im( conj(Uq[m,i]) * Uk[m,j] )
    }
    Wr[idx] = wr; Wi[idx] = wi;
    Uvr[idx] = Uv[idx * 2 + 0];
    Uvi[idx] = Uv[idx * 2 + 1];
}

// ---------------------------------------------------------------------------
// T_m = L * X_m (complex), m = 0..31.  One wave per 16x16 output tile.
// Tr = Lr Xr - Li Xi ; Ti = Li Xr + Lr Xi.
__global__ void k_lmul(const float* __restrict__ Lr, const float* __restrict__ Li,
                       const float* __restrict__ Xr, const float* __restrict__ Xi,
                       float* __restrict__ Tr, float* __restrict__ Ti) {
    int wave = blockIdx.x * (blockDim.x >> 5) + (threadIdx.x >> 5);
    int lane = threadIdx.x & 31;
    int m    = wave >> 6;            // matrix index
    int tile = wave & 63;            // 8x8 tiles of 16x16
    int i0 = (tile >> 3) << 4;
    int j0 = (tile & 7) << 4;
    const float* xr = Xr + m * DD;
    const float* xi = Xi + m * DD;
    int arow = i0 + (lane & 15);     // A operand: M = lane&15
    int bcol = j0 + (lane & 15);     // B operand: N = lane&15
    int ksub = (lane >> 4) << 1;     // K sub-block: 0 or 2
    v8f accr = {}; v8f acci = {};
    for (int k0 = 0; k0 < DDIM; k0 += 4) {
        int ka = k0 + ksub;
        v2f ar  = *(const v2f*)(Lr + arow * DDIM + ka);
        v2f ai  = *(const v2f*)(Li + arow * DDIM + ka);
        v2f nai = -ai;
        v2f br, bi;
        br.x = xr[ka * DDIM + bcol];       br.y = xr[(ka + 1) * DDIM + bcol];
        bi.x = xi[ka * DDIM + bcol];       bi.y = xi[(ka + 1) * DDIM + bcol];
        accr = wmma_f32(ar,  br, accr);    // + Lr Xr
        accr = wmma_f32(nai, bi, accr);    // - Li Xi
        acci = wmma_f32(ai,  br, acci);    // + Li Xr
        acci = wmma_f32(ar,  bi, acci);    // + Lr Xi
    }
    int mrow = i0 + ((lane >> 4) << 3);    // C/D: VGPR v -> M = v + (lane>>4)*8
    int col  = j0 + (lane & 15);
    float* trp = Tr + m * DD;
    float* tip = Ti + m * DD;
    for (int v = 0; v < 8; ++v) {
        trp[(mrow + v) * DDIM + col] = accr[v];
        tip[(mrow + v) * DDIM + col] = acci[v];
    }
}

// ---------------------------------------------------------------------------
// O_m = T_m * L^H :  Or = Tr Lr^T + Ti Li^T ; Oi = Ti Lr^T - Tr Li^T.
// B operand element B[k,n] = L[(j0+n)*D + k] -> contiguous float2 loads.
// transpose_out=1 stores O transposed (for coalesced trace reads).
__global__ void k_rmulh(const float* __restrict__ Tr, const float* __restrict__ Ti,
                        const float* __restrict__ Lr, const float* __restrict__ Li,
                        float* __restrict__ Or, float* __restrict__ Oi,
                        int transpose_out) {
    int wave = blockIdx.x * (blockDim.x >> 5) + (threadIdx.x >> 5);
    int lane = threadIdx.x & 31;
    int m    = wave >> 6;
    int tile = wave & 63;
    int i0 = (tile >> 3) << 4;
    int j0 = (tile & 7) << 4;
    const float* trp = Tr + m * DD;
    const float* tip = Ti + m * DD;
    int arow = i0 + (lane & 15);
    int brow = j0 + (lane & 15);     // row of L (we need L[j, k])
    int ksub = (lane >> 4) << 1;
    v8f orr = {}; v8f oii = {};
    for (int k0 = 0; k0 < DDIM; k0 += 4) {
        int ka = k0 + ksub;
        v2f atr  = *(const v2f*)(trp + arow * DDIM + ka);
        v2f ati  = *(const v2f*)(tip + arow * DDIM + ka);
        v2f natr = -atr;
        v2f br   = *(const v2f*)(Lr + brow * DDIM + ka);
        v2f bi   = *(const v2f*)(Li + brow * DDIM + ka);
        orr = wmma_f32(atr,  br, orr);     // + Tr Lr^T
        orr = wmma_f32(ati,  bi, orr);     // + Ti Li^T
        oii = wmma_f32(ati,  br, oii);     // + Ti Lr^T
        oii = wmma_f32(natr, bi, oii);     // - Tr Li^T
    }
    int mrow = i0 + ((lane >> 4) << 3);
    int col  = j0 + (lane & 15);
    float* orp = Or + m * DD;
    float* oip = Oi + m * DD;
    if (transpose_out) {
        for (int v = 0; v < 8; ++v) {
            orp[col * DDIM + (mrow + v)] = orr[v];
            oip[col * DDIM + (mrow + v)] = oii[v];
        }
    } else {
        for (int v = 0; v < 8; ++v) {
            orp[(mrow + v) * DDIM + col] = orr[v];
            oip[(mrow + v) * DDIM + col] = oii[v];
        }
    }
}

// ---------------------------------------------------------------------------
// trace[t,k,b] = sum_{i,j} X[t,b,i,j] * A[k,b,j,i]  (complex).
// A is stored transposed (AT[i*D+j] = A[j*D+i]) so both streams are coalesced.
// One block per (t,b); this kernel streams the full 64MB of inputs.
__global__ void k_trace(const float* __restrict__ Xr, const float* __restrict__ Xi,
                        const float* __restrict__ ATr, const float* __restrict__ ATi,
                        float* __restrict__ trr, float* __restrict__ tri) {
    int m = blockIdx.x;          // t*16 + b
    int b = m & 15;
    int tid = threadIdx.x;
    const float4* xr4 = (const float4*)(Xr + m * DD);
    const float4* xi4 = (const float4*)(Xi + m * DD);
    __shared__ float2 red[256];
    for (int k = 0; k < 2; ++k) {
        const float4* ar4 = (const float4*)(ATr + (k * BBATCH + b) * DD);
        const float4* ai4 = (const float4*)(ATi + (k * BBATCH + b) * DD);
        float sr = 0.f, si = 0.f;
        for (int s = tid; s < DD / 4; s += blockDim.x) {
            float4 xr = xr4[s], xi = xi4[s], ar = ar4[s], ai = ai4[s];
            sr += xr.x * ar.x - xi.x * ai.x;  si += xr.x * ai.x + xi.x * ar.x;
            sr += xr.y * ar.y - xi.y * ai.y;  si += xr.y * ai.y + xi.y * ar.y;
            sr += xr.z * ar.z - xi.z * ai.z;  si += xr.z * ai.z + xi.z * ar.z;
            sr += xr.w * ar.w - xi.w * ai.w;  si += xr.w * ai.w + xi.w * ar.w;
        }
        red[tid].x = sr; red[tid].y = si;
        __syncthreads();
        for (int off = 128; off > 0; off >>= 1) {
            if (tid < off) { red[tid].x += red[tid + off].x; red[tid].y += red[tid + off].y; }
            __syncthreads();
        }
        if (tid == 0) { trr[m * 2 + k] = red[0].x; tri[m * 2 + k] = red[0].y; }
        __syncthreads();
    }
}

// ---------------------------------------------------------------------------
// dist = |trace|; softmax over b (16) for each (q, k in {0,1}).
__global__ void k_softmax(const float* __restrict__ trr, const float* __restrict__ tri,
                          float* __restrict__ s0, float* __restrict__ s1) {
    int tid = threadIdx.x;
    if (tid >= 64) return;
    int q = tid >> 1, k = tid & 1;
    float d[16];
    float mx = -1e30f;
    for (int b = 0; b < 16; ++b) {
        int m = q * 16 + b;
        float r = trr[m * 2 + k], i = tri[m * 2 + k];
        d[b] = sqrtf(r * r + i * i);
        mx = fmaxf(mx, d[b]);
    }
    float sum = 0.f;
    for (int b = 0; b < 16; ++b) { d[b] = __expf(d[b] - mx); sum += d[b]; }
    float inv = 1.f / sum;
    float* sp = (k == 0) ? s0 : s1;
    for (int b = 0; b < 16; ++b) sp[q * 16 + b] = d[b] * inv;
}

// ---------------------------------------------------------------------------
// out_r[t,b] = Vr(t=0,b)*s0 + Vi(t=0,b)*s1 ; out_i[t,b] = Vr(t=1,b)*s0 + Vi(t=1,b)*s1.
// Writes the 64MB output; V matrices (4MB) stay L2-resident.
__global__ void k_out(const float* __restrict__ Vr, const float* __restrict__ Vi,
                      const float* __restrict__ s0, const float* __restrict__ s1,
                      float* __restrict__ out) {
    int m = blockIdx.x;          // t*16 + b
    int b = m & 15;
    float a0 = s0[m], a1 = s1[m];
    const float4* vr0 = (const float4*)(Vr + (0 * BBATCH + b) * DD);
    const float4* vi0 = (const float4*)(Vi + (0 * BBATCH + b) * DD);
    const float4* vr1 = (const float4*)(Vr + (1 * BBATCH + b) * DD);
    const float4* vi1 = (const float4*)(Vi + (1 * BBATCH + b) * DD);
    float4* or4 = (float4*)(out + (size_t)m * DD);
    float4* oi4 = (float4*)(out + (size_t)TT * BBATCH * DD + (size_t)m * DD);
    for (int s = threadIdx.x; s < DD / 4; s += blockDim.x) {
        float4 r0 = vr0[s], i0 = vi0[s], r1 = vr1[s], i1 = vi1[s];
        float4 o_r, o_i;
        o_r.x = r0.x * a0 + i0.x * a1;  o_i.x = r1.x * a0 + i1.x * a1;
        o_r.y = r0.y * a0 + i0.y * a1;  o_i.y = r1.y * a0 + i1.y * a1;
        o_r.z = r0.z * a0 + i0.z * a1;  o_i.z = r1.z * a0 + i1.z * a1;
        o_r.w = r0.w * a0 + i0.w * a1;  o_i.w = r1.w * a0 + i1.w * a1;
        or4[s] = o_r;
        oi4[s] = o_i;
    }
}

// ---------------------------------------------------------------------------
extern "C" void kernel_launch(void* const* d_in, const int* in_sizes, int n_in,
                              void* d_out, int out_size, void* d_ws, size_t ws_size,
                              hipStream_t stream) {
    const float* xr = (const float*)d_in[0];   // inputs_real (T,B,D,D)
    const float* xi = (const float*)d_in[1];   // inputs_imag (T,B,D,D)
    const float* Uq = (const float*)d_in[2];   // (D,D,2)
    const float* Uk = (const float*)d_in[3];
    const float* Uv = (const float*)d_in[4];
    float* out = (float*)d_out;
    float* ws  = (float*)d_ws;

    // Workspace layout (floats); total ~3.2M floats (~13 MB).
    float* Wr  = ws;
    float* Wi  = Wr  + DD;
    float* Uvr = Wi  + DD;
    float* Uvi = Uvr + DD;
    float* Tr  = Uvi + DD;          // temp left-product, 32 matrices
    float* Ti  = Tr  + NM * DD;
    float* ATr = Ti  + NM * DD;     // A = W X W^H, stored transposed
    float* ATi = ATr + NM * DD;
    float* Vr  = ATi + NM * DD;     // V = Uv X Uv^H for t in {0,1}
    float* Vi  = Vr  + NM * DD;
    float* trr = Vi  + NM * DD;     // 1024 complex traces
    float* tri = trr + TT * BBATCH * 2;
    float* s0  = tri + TT * BBATCH * 2;
    float* s1  = s0  + TT * BBATCH;

    // 1) W = Uq^H Uk, de-interleave Uv.
    k_prep<<<64, 256, 0, stream>>>(Uq, Uk, Uv, Wr, Wi, Uvr, Uvi);
    // 2) A_{k,b} = W X_{k,b} W^H  (X at t=0,1 are the first 32 input matrices).
    k_lmul <<<512, 128, 0, stream>>>(Wr, Wi, xr, xi, Tr, Ti);
    k_rmulh<<<512, 128, 0, stream>>>(Tr, Ti, Wr, Wi, ATr, ATi, 1);
    // 3) V_{t,b} = Uv X_{t,b} Uv^H for t=0,1.
    k_lmul <<<512, 128, 0, stream>>>(Uvr, Uvi, xr, xi, Tr, Ti);
    k_rmulh<<<512, 128, 0, stream>>>(Tr, Ti, Uvr, Uvi, Vr, Vi, 0);
    // 4) Complex traces tr(X_{t,b} A_{k,b}) for k=0,1 (streams all inputs).
    k_trace<<<512, 256, 0, stream>>>(xr, xi, ATr, ATi, trr, tri);
    // 5) softmax over b.
    k_softmax<<<1, 64, 0, stream>>>(trr, tri, s0, s1);
    // 6) Scaled-V output write (64 MB).
    k_out<<<512, 256, 0, stream>>>(Vr, Vi, s0, s1, out);
}